// GCN_55027120996899
// MI455X (gfx1250) — compile-verified
//
#include <hip/hip_runtime.h>

#define NODES  50000
#define EDGES  800000
#define LEDGES 200000
#define INCH   256
#define NF     64

typedef float v2f __attribute__((ext_vector_type(2)));
typedef float v8f __attribute__((ext_vector_type(8)));

// ---------------- degree / norm ----------------
__global__ void k_deg_init(float* deg, int n) {
  int i = blockIdx.x * blockDim.x + threadIdx.x;
  if (i < n) deg[i] = 1.0f;               // self-loop
}

__global__ void k_deg_accum(const int* __restrict__ col, float* deg, int e) {
  int i = blockIdx.x * blockDim.x + threadIdx.x;
  if (i < e) atomicAdd(&deg[col[i]], 1.0f);
}

__global__ void k_deg_rsqrt(float* deg, int n) {
  int i = blockIdx.x * blockDim.x + threadIdx.x;
  if (i < n) deg[i] = rsqrtf(deg[i]);     // deg >= 1 always
}

// ---------------- WMMA GEMM: [M,K] x [K,64] -> [M,64], f32 ----------------
// Block = 256 threads = 8 waves. Each wave computes a 16-row strip across all
// 64 output columns via 4 accumulator tiles of V_WMMA_F32_16X16X4_F32.
// W is staged into LDS pre-swizzled into per-lane B fragments so the inner
// loop does one ds_load_b64 per tile (no register shuffling between WMMAs).
template<int K>
__global__ __launch_bounds__(256) void k_gemm(const float* __restrict__ A,
                                              const float* __restrict__ W,
                                              float* __restrict__ C, int M) {
  // 64-row chunk of W in fragment order: [kstep(16)][tile(4)][lane(32)] x float2
  __shared__ __align__(16) float sB[4096];   // 16 KB

  const int lane = threadIdx.x & 31;
  const int wave = threadIdx.x >> 5;

  int strip = blockIdx.x * 8 + wave;      // wave-uniform
  int maxStrip = (M >> 4) - 1;            // M divisible by 16
  if (strip > maxStrip) strip = maxStrip; // clamp: keeps EXEC all-1 for WMMA
  const long rowBase = (long)strip << 4;

  const int mlo  = lane & 15;             // M (for A) / N (for B,C) low index
  const int koff = (lane >> 4) << 1;      // lanes 16-31 hold K+2,K+3 of A frag
  const float* arow = A + (rowBase + mlo) * (long)K;

  v8f acc[4];
#pragma unroll
  for (int t = 0; t < 4; ++t) acc[t] = (v8f){0.f,0.f,0.f,0.f,0.f,0.f,0.f,0.f};

  for (int k0 = 0; k0 < K; k0 += 64) {
    __syncthreads();
    // stage + swizzle W chunk into fragment order
    for (int idx = threadIdx.x; idx < 2048; idx += 256) {
      int ks = idx >> 7;                  // k-step 0..15
      int t  = (idx >> 5) & 3;            // tile
      int l  = idx & 31;                  // lane slot
      int kk = k0 + ks * 4 + ((l >> 4) << 1);
      int cl = t * 16 + (l & 15);
      sB[idx * 2 + 0] = W[kk * NF + cl];
      sB[idx * 2 + 1] = W[(kk + 1) * NF + cl];
    }
    __syncthreads();
    if (k0 + 64 < K)                      // prefetch next A chunk
      __builtin_prefetch(arow + k0 + 64, 0, 0);
#pragma unroll 4
    for (int ks = 0; ks < 16; ++ks) {
      // A fragment: one b64 load per lane (K=0..1 lanes 0-15, K=2..3 lanes 16-31)
      v2f a = *(const v2f*)(arow + k0 + ks * 4 + koff);
      const v2f* bfrag = (const v2f*)sB + ks * 128 + lane;
#pragma unroll
      for (int t = 0; t < 4; ++t) {
        v2f b = bfrag[t * 32];            // single ds_load_b64, imm offset
        acc[t] = __builtin_amdgcn_wmma_f32_16x16x4_f32(
            false, a, false, b, (short)0, acc[t], false, false);
      }
    }
  }

  // C/D layout: VGPR i -> row (mhi + i), lanes 0-15 N=n, lanes 16-31 rows 8..15
  const int mhi = (lane >> 4) << 3;
#pragma unroll
  for (int t = 0; t < 4; ++t)
#pragma unroll
    for (int i = 0; i < 8; ++i)
      C[(rowBase + mhi + i) * NF + t * 16 + mlo] = acc[t][i];
}

// ---------------- aggregation ----------------
// 16 threads per node, each handles 4 features.
__global__ void k_agg_init(const float* __restrict__ xw, const float* __restrict__ dinv,
                           float* __restrict__ agg, int n) {
  int t = blockIdx.x * blockDim.x + threadIdx.x;
  int node = t >> 4;
  int f = (t & 15) << 2;
  if (node < n) {
    float s = dinv[node]; s = s * s;        // self-loop norm
    float4 v = *(const float4*)(xw + (long)node * NF + f);
    v.x *= s; v.y *= s; v.z *= s; v.w *= s;
    *(float4*)(agg + (long)node * NF + f) = v;
  }
}

// 16 threads per edge, each gathers float4 and scatters 4 float atomics.
__global__ void k_agg_edges(const float* __restrict__ xw, const int* __restrict__ row,
                            const int* __restrict__ col, const float* __restrict__ dinv,
                            float* agg, int e) {
  int t = blockIdx.x * blockDim.x + threadIdx.x;
  int ed = t >> 4;
  int f = (t & 15) << 2;
  if (ed < e) {
    int r = row[ed], c = col[ed];
    float nrm = dinv[r] * dinv[c];
    float4 v = *(const float4*)(xw + (long)r * NF + f);
    float* dst = agg + (long)c * NF + f;
    atomicAdd(dst + 0, v.x * nrm);
    atomicAdd(dst + 1, v.y * nrm);
    atomicAdd(dst + 2, v.z * nrm);
    atomicAdd(dst + 3, v.w * nrm);
  }
}

template<bool RELU>
__global__ void k_bias_act(const float* __restrict__ agg, const float* __restrict__ bias,
                           float* __restrict__ out, int n) {
  int t = blockIdx.x * blockDim.x + threadIdx.x;
  int node = t >> 4;
  int f = (t & 15) << 2;
  if (node < n) {
    float4 v = *(const float4*)(agg + (long)node * NF + f);
    v.x += bias[f + 0]; v.y += bias[f + 1]; v.z += bias[f + 2]; v.w += bias[f + 3];
    if (RELU) {
      v.x = fmaxf(v.x, 0.f); v.y = fmaxf(v.y, 0.f);
      v.z = fmaxf(v.z, 0.f); v.w = fmaxf(v.w, 0.f);
    }
    *(float4*)(out + (long)node * NF + f) = v;
  }
}

// ---------------- decoder: dot(z[src], z[dst]) ----------------
__global__ void k_decode(const float* __restrict__ z, const int* __restrict__ src,
                         const int* __restrict__ dst, float* __restrict__ out, int e) {
  int i = blockIdx.x * blockDim.x + threadIdx.x;
  if (i < e) {
    const float4* a = (const float4*)(z + (long)src[i] * NF);
    const float4* b = (const float4*)(z + (long)dst[i] * NF);
    float acc = 0.f;
#pragma unroll
    for (int j = 0; j < 16; ++j) {
      float4 av = a[j], bv = b[j];
      acc += av.x * bv.x + av.y * bv.y + av.z * bv.z + av.w * bv.w;
    }
    out[i] = acc;
  }
}

// ---------------- launch ----------------
extern "C" void kernel_launch(void* const* d_in, const int* in_sizes, int n_in,
                              void* d_out, int out_size, void* d_ws, size_t ws_size,
                              hipStream_t stream) {
  const float* x   = (const float*)d_in[0];
  const int*   ei  = (const int*)d_in[1];
  const int*   eli = (const int*)d_in[2];
  const float* W1  = (const float*)d_in[3];
  const float* b1  = (const float*)d_in[4];
  const float* W2  = (const float*)d_in[5];
  const float* b2  = (const float*)d_in[6];
  float* out = (float*)d_out;

  float* ws   = (float*)d_ws;
  float* dinv = ws;                                  // 50000 (+48 pad)
  float* bufA = ws + 50048;                          // 3.2M floats
  float* bufB = bufA + (size_t)NODES * NF;
  float* bufC = bufB + (size_t)NODES * NF;

  const int* erow = ei;
  const int* ecol = ei + EDGES;
  const int* lsrc = eli;
  const int* ldst = eli + LEDGES;

  dim3 blk(256);
  const int nodeGrid = (NODES * 16 + 255) / 256;     // node x feat4
  const int edgeGrid = (EDGES * 16 + 255) / 256;     // edge x feat4
  const int strips   = NODES / 16;                   // 3125
  const int gemmGrid = (strips + 7) / 8;             // 8 waves per block

  // normalization
  k_deg_init<<<(NODES + 255) / 256, blk, 0, stream>>>(dinv, NODES);
  k_deg_accum<<<(EDGES + 255) / 256, blk, 0, stream>>>(ecol, dinv, EDGES);
  k_deg_rsqrt<<<(NODES + 255) / 256, blk, 0, stream>>>(dinv, NODES);

  // layer 1: xw1 = x @ W1 ; agg ; h = relu(agg + b1)
  k_gemm<INCH><<<gemmGrid, blk, 0, stream>>>(x, W1, bufA, NODES);
  k_agg_init<<<nodeGrid, blk, 0, stream>>>(bufA, dinv, bufB, NODES);
  k_agg_edges<<<edgeGrid, blk, 0, stream>>>(bufA, erow, ecol, dinv, bufB, EDGES);
  k_bias_act<true><<<nodeGrid, blk, 0, stream>>>(bufB, b1, bufA, NODES);

  // layer 2: hw2 = h @ W2 ; agg ; z = agg + b2
  k_gemm<NF><<<gemmGrid, blk, 0, stream>>>(bufA, W2, bufB, NODES);
  k_agg_init<<<nodeGrid, blk, 0, stream>>>(bufB, dinv, bufC, NODES);
  k_agg_edges<<<edgeGrid, blk, 0, stream>>>(bufB, erow, ecol, dinv, bufC, EDGES);
  k_bias_act<false><<<nodeGrid, blk, 0, stream>>>(bufC, b2, bufA, NODES);

  // decode
  k_decode<<<(LEDGES + 255) / 256, blk, 0, stream>>>(bufA, lsrc, ldst, out, LEDGES);
}